// Sdf2DensityRayMarcher_31748398252671
// MI455X (gfx1250) — compile-verified
//
#include <hip/hip_runtime.h>
#include <stdint.h>

// Sdf2DensityRayMarcher — MI455X (gfx1250, wave32)
// colors [4,16384,48,32] f32, sdfs [4,16384,48,1] f32, depths [4,16384,48,1] f32 (sorted)
// out = concat( rgb [B,R,32], depth [B,R,1], weights [B,R,48,1] )  (flat f32)
// Bandwidth-bound (~450 MB @ 23.3 TB/s ≈ 19 us). One wave per ray; async HBM->LDS
// streaming of the per-ray 6 KB color tile (CDNA5 ASYNCcnt path) with NT hint
// (402 MB stream-once > 192 MB L2 -> don't cache it).

#define BETA 0.02f
static constexpr int Bb = 4, Rr = 16384, Ss = 48, Cc = 32;
static constexpr int NRAYS = Bb * Rr;                  // 65536

// ---------- monotonic float<->uint encoding for global min/max ----------
__device__ __forceinline__ unsigned enc_f32(float f) {
  unsigned u = __float_as_uint(f);
  return (u & 0x80000000u) ? ~u : (u | 0x80000000u);
}
__device__ __forceinline__ float dec_f32(unsigned e) {
  unsigned u = (e & 0x80000000u) ? (e & 0x7FFFFFFFu) : ~e;
  return __uint_as_float(u);
}

// ---------- CDNA5 async HBM->LDS copy (per-lane b128) ----------
// Builtin signature (probe-confirmed): (int4 __device__* src, int4 __shared__* dst,
// imm offset, imm cpol). cpol = 1 -> TH_NT (non-temporal: stream-once data).
typedef int v4i __attribute__((ext_vector_type(4)));
typedef __attribute__((address_space(1))) v4i gv4i;   // global (prints as __device__)
typedef __attribute__((address_space(3))) v4i lv4i;   // LDS    (prints as __shared__)

__device__ __forceinline__ void async_copy_b128_nt(const void* g, void* l) {
#if __has_builtin(__builtin_amdgcn_global_load_async_to_lds_b128)
  __builtin_amdgcn_global_load_async_to_lds_b128((gv4i*)g, (lv4i*)l, 0, /*TH_NT*/1);
#else
  unsigned lds_addr = (unsigned)(uintptr_t)l;  // generic->LDS: low 32 bits are LDS offset
  unsigned long long gaddr = (unsigned long long)(uintptr_t)g;
  asm volatile("global_load_async_to_lds_b128 %0, %1, off th:TH_LOAD_NT"
               :: "v"(lds_addr), "v"(gaddr) : "memory");
#endif
}

__device__ __forceinline__ void wait_async0() {
#if __has_builtin(__builtin_amdgcn_s_wait_asynccnt)
  __builtin_amdgcn_s_wait_asynccnt(0);
#else
  asm volatile("s_wait_asynccnt 0" ::: "memory");
#endif
  asm volatile("" ::: "memory");  // keep LDS reads after the wait
}

// ---------- main: one wave32 per ray, lane == channel (first in file so the
// ---------- disasm snippet shows the async-tensor instructions) ----------
__global__ void __launch_bounds__(256)
raymarch_kernel(const float* __restrict__ colors,
                const float* __restrict__ sdfs,
                const float* __restrict__ depths,
                float* __restrict__ out_rgb,
                float* __restrict__ out_depth,
                float* __restrict__ out_w,
                const unsigned* __restrict__ mm) {
  __shared__ float s_col[8][Ss * Cc];   // 8 * 6144 B = 48 KB
  __shared__ float s_depth[8][Ss];
  __shared__ float s_alpha[8][Ss];
  __shared__ float s_w[8][Ss];

  const int wv   = threadIdx.x >> 5;
  const int lane = threadIdx.x & 31;
  const int ray  = blockIdx.x * 8 + wv;

  const size_t cbase = (size_t)ray * (Ss * Cc);
  const int    sbase = ray * Ss;

  // Kick off async streaming of this ray's 6144-B color tile: 12 x (32 lanes x 16 B).
  {
    const char* g = (const char*)(colors + cbase) + lane * 16;
    char*       l = (char*)&s_col[wv][0] + lane * 16;
    #pragma unroll
    for (int i = 0; i < 12; ++i)
      async_copy_b128_nt(g + i * 512, l + i * 512);
  }

  // While DMA streams colors: depths -> LDS, per-sample alpha (2 exps/sample).
  float d_own = depths[sbase + lane];        // s = lane
  s_depth[wv][lane] = d_own;
  float d_hi = 0.0f;
  if (lane < 16) {                           // s = 32 + lane
    d_hi = depths[sbase + 32 + lane];
    s_depth[wv][32 + lane] = d_hi;
  }
  __builtin_amdgcn_wave_barrier();           // intra-wave LDS is in-order; sched fence only

  {
    float sd    = sdfs[sbase + lane];
    float dens  = 1.0f / (BETA * (1.0f + __expf(sd * (1.0f / BETA))));  // sigmoid(-sd/b)/b
    float delta = s_depth[wv][lane + 1] - d_own;   // lane 31 reads s_depth[32] (valid)
    s_alpha[wv][lane] = 1.0f - __expf(-dens * delta);
  }
  if (lane < 16) {
    int   s     = 32 + lane;
    float sd    = sdfs[sbase + s];
    float dens  = 1.0f / (BETA * (1.0f + __expf(sd * (1.0f / BETA))));
    float delta = (s < Ss - 1) ? (s_depth[wv][s + 1] - d_hi) : 1e9f;
    s_alpha[wv][s] = 1.0f - __expf(-dens * delta);
  }
  __builtin_amdgcn_wave_barrier();

  wait_async0();                             // colors resident in LDS

  // Fused transmittance recurrence + weighted composite. Broadcast LDS reads for
  // alpha/depth (uniform across lanes), conflict-free s_col reads (32 consecutive f32).
  float T = 1.0f, rgb = 0.0f, wsum = 0.0f, dsum = 0.0f;
  #pragma unroll 8
  for (int s = 0; s < Ss; ++s) {
    float a = s_alpha[wv][s];
    float w = a * T;
    rgb  = fmaf(w, s_col[wv][s * Cc + lane], rgb);
    wsum += w;
    dsum = fmaf(w, s_depth[wv][s], dsum);
    if (lane == 0) s_w[wv][s] = w;
    T *= (1.0f - a + 1e-10f);
  }
  __builtin_amdgcn_wave_barrier();

  // white background + [-1,1] rescale
  out_rgb[ray * Cc + lane] = (rgb + 1.0f - wsum) * 2.0f - 1.0f;

  // weights out, coalesced b32 across the wave
  out_w[sbase + lane] = s_w[wv][lane];
  if (lane < 16) out_w[sbase + 32 + lane] = s_w[wv][32 + lane];

  if (lane == 0) {
    float dmin  = dec_f32(mm[0]);
    float dmax  = dec_f32(mm[1]);
    float dlast = s_depth[wv][Ss - 1];
    float wd    = dsum + (1.0f - wsum) * dlast;    // weights.at[-1].add(1-wsum)
    float cd    = wd / wsum;
    if (isnan(cd)) cd = dmax;                      // nan->inf->clip == dmax
    cd = fminf(fmaxf(cd, dmin), dmax);             // fminf(inf,dmax)=dmax
    out_depth[ray] = cd;
  }
}

__global__ void init_mm_kernel(unsigned* mm) {
  if (threadIdx.x == 0) { mm[0] = 0xFFFFFFFFu; mm[1] = 0u; }  // min-acc, max-acc
}

__global__ void depth_minmax_kernel(const float* __restrict__ depths,
                                    unsigned* __restrict__ mm, int n) {
  unsigned lmin = 0xFFFFFFFFu, lmax = 0u;
  for (int i = blockIdx.x * blockDim.x + threadIdx.x; i < n;
       i += gridDim.x * blockDim.x) {
    unsigned e = enc_f32(depths[i]);
    lmin = lmin < e ? lmin : e;
    lmax = lmax > e ? lmax : e;
  }
  for (int off = 16; off > 0; off >>= 1) {
    unsigned om = (unsigned)__shfl_down((int)lmin, off, 32);
    unsigned oM = (unsigned)__shfl_down((int)lmax, off, 32);
    lmin = lmin < om ? lmin : om;
    lmax = lmax > oM ? lmax : oM;
  }
  __shared__ unsigned smin[8], smax[8];
  const int wv = threadIdx.x >> 5, ln = threadIdx.x & 31;
  if (ln == 0) { smin[wv] = lmin; smax[wv] = lmax; }
  __syncthreads();
  if (threadIdx.x == 0) {
    unsigned m = smin[0], M = smax[0];
    #pragma unroll
    for (int i = 1; i < 8; ++i) { m = min(m, smin[i]); M = max(M, smax[i]); }
    atomicMin(&mm[0], m);
    atomicMax(&mm[1], M);
  }
}

extern "C" void kernel_launch(void* const* d_in, const int* in_sizes, int n_in,
                              void* d_out, int out_size, void* d_ws, size_t ws_size,
                              hipStream_t stream) {
  const float* colors = (const float*)d_in[0];
  const float* sdfs   = (const float*)d_in[1];
  const float* depths = (const float*)d_in[2];

  float* out       = (float*)d_out;
  float* out_rgb   = out;                                  // B*R*C
  float* out_depth = out_rgb + (size_t)NRAYS * Cc;         // B*R
  float* out_w     = out_depth + NRAYS;                    // B*R*S
  unsigned* mm     = (unsigned*)d_ws;                      // [min_enc, max_enc]

  init_mm_kernel<<<1, 32, 0, stream>>>(mm);
  depth_minmax_kernel<<<1024, 256, 0, stream>>>(depths, mm, NRAYS * Ss);
  raymarch_kernel<<<NRAYS / 8, 256, 0, stream>>>(colors, sdfs, depths,
                                                 out_rgb, out_depth, out_w, mm);
}